// Net_51032801411773
// MI455X (gfx1250) — compile-verified
//
#include <hip/hip_runtime.h>

typedef float v2f __attribute__((ext_vector_type(2)));
typedef float v8f __attribute__((ext_vector_type(8)));

// ---------------------------------------------------------------------------
// Phase 1: xgT[t][i][k] = sum_j x[t,j]*W_ih[k*10+i, j] + (b_ih+b_hh)[k*10+i]
// (gate-interleaved layout so the scan reads one float4 per step per lane).
// GEMM via V_WMMA_F32_16X16X4_F32: one wave = 16 timesteps, K=10 padded to
// 3 chunks of 4, N=40 as 3 tiles of 16 (cols >=40 masked at store).
// A (16x4 f32): lanes 0-15 -> M=lane, {K0,K1}; lanes 16-31 -> {K2,K3}.
// B (4x16 f32): lanes index N; VGPR0={K0|K2}, VGPR1={K1|K3}.
// C/D (16x16):  VGPR r -> M=r (lanes 0-15) / M=r+8 (lanes 16-31).
// ---------------------------------------------------------------------------
__global__ void __launch_bounds__(128)
xg_wmma_kernel(const float* __restrict__ x,
               const float* __restrict__ W_ih,
               const float* __restrict__ b_ih,
               const float* __restrict__ b_hh,
               float* __restrict__ xg,      // gate-interleaved [T][10][4]
               int T, int ntiles)
{
    const int lane = threadIdx.x & 31;
    const int wave = threadIdx.x >> 5;
    const int tile = blockIdx.x * 4 + wave;
    if (tile >= ntiles) return;

    const int t0 = tile * 16;
    const int m  = lane & 15;
    const int hi = lane >> 4;

    // A fragments: rows of x (row-clamped; clamped rows reproduce row T-1,
    // making their later clamped stores benign duplicates).
    int row = t0 + m; if (row > T - 1) row = T - 1;
    const float* xr = x + (size_t)row * 10;
    v2f a[3];
#pragma unroll
    for (int kc = 0; kc < 3; ++kc) {
        int k0  = kc * 4 + hi * 2;
        int k0c = (k0     < 10) ? k0     : 0;
        int k1c = (k0 + 1 < 10) ? k0 + 1 : 0;
        float va = xr[k0c], vb = xr[k1c];
        a[kc].x = (k0     < 10) ? va : 0.0f;
        a[kc].y = (k0 + 1 < 10) ? vb : 0.0f;
    }

#pragma unroll
    for (int nt = 0; nt < 3; ++nt) {
        int col  = nt * 16 + m;          // gate row index 0..39 (masked >=40)
        int colc = (col < 40) ? col : 39;
        float bsum = b_ih[colc] + b_hh[colc];
        float bias = (col < 40) ? bsum : 0.0f;

        v8f acc;
#pragma unroll
        for (int r = 0; r < 8; ++r) acc[r] = bias;

#pragma unroll
        for (int kc = 0; kc < 3; ++kc) {
            int k0  = kc * 4 + hi * 2;
            int k0c = (k0     < 10) ? k0     : 0;
            int k1c = (k0 + 1 < 10) ? k0 + 1 : 0;
            float w0 = W_ih[(size_t)colc * 10 + k0c];
            float w1 = W_ih[(size_t)colc * 10 + k1c];
            v2f b;
            b.x = (col < 40 && k0     < 10) ? w0 : 0.0f;
            b.y = (col < 40 && k0 + 1 < 10) ? w1 : 0.0f;
            acc = __builtin_amdgcn_wmma_f32_16x16x4_f32(
                false, a[kc], false, b, (short)0, acc, false, false);
        }

        // Gate-interleaved store: offset = trow*40 + (col%10)*4 + col/10.
        // Single loop-invariant mask -> one exec region, 8 plain stores.
        int kk = colc / 10;              // gate 0..3
        int ii = colc - kk * 10;         // element 0..9
        size_t coff = (size_t)ii * 4 + kk;
        if (col < 40) {
#pragma unroll
            for (int r = 0; r < 8; ++r) {
                int trow = t0 + r + hi * 8;
                if (trow > T - 1) trow = T - 1;     // clamped duplicate store
                xg[(size_t)trow * 40 + coff] = acc[r];
            }
        }
    }
}

// ---------------------------------------------------------------------------
// Phase 2: latency-bound sequential LSTM scan, single wave32.
// ---------------------------------------------------------------------------
__device__ __forceinline__ float bcastf(float v, int l) {
    return __builtin_bit_cast(float,
        __builtin_amdgcn_readlane(__builtin_bit_cast(int, v), l));
}

__device__ __forceinline__ float fsigmoid(float xx) {
    float e = __builtin_amdgcn_exp2f(xx * -1.44269504088896341f);
    return __builtin_amdgcn_rcpf(1.0f + e);
}

__device__ __forceinline__ float ftanhf(float xx) {
#if __has_builtin(__builtin_amdgcn_tanhf)
    return __builtin_amdgcn_tanhf(xx);
#else
    float e = __builtin_amdgcn_exp2f(xx * -2.88539008177792681f);
    return fmaf(2.0f, __builtin_amdgcn_rcpf(1.0f + e), -1.0f);
#endif
}

// 10-wide dot, two chains (depth ~5) to shorten the critical path.
__device__ __forceinline__ float dot10(const float* hs, const float* w, float acc) {
    float a = fmaf(hs[8], w[8], acc);
    a = fmaf(hs[6], w[6], a);
    a = fmaf(hs[4], w[4], a);
    a = fmaf(hs[2], w[2], a);
    a = fmaf(hs[0], w[0], a);
    float b = hs[9] * w[9];
    b = fmaf(hs[7], w[7], b);
    b = fmaf(hs[5], w[5], b);
    b = fmaf(hs[3], w[3], b);
    b = fmaf(hs[1], w[1], b);
    return a + b;
}

#ifndef SCAN_PF
#define SCAN_PF 8     // software-pipeline depth (prefetch distance in steps)
#endif

template <bool FUSED>
__global__ void __launch_bounds__(32)
lstm_scan_kernel(const float* __restrict__ src,   // FUSED ? x[T,10] : xgT[T,10,4]
                 const float* __restrict__ W_hh,
                 const float* __restrict__ W_ih,  // FUSED only
                 const float* __restrict__ b_ih,  // FUSED only
                 const float* __restrict__ b_hh,  // FUSED only
                 const float* __restrict__ h0,
                 const float* __restrict__ c0,
                 float* __restrict__ out, int T)
{
    const int lane = threadIdx.x;
    const int li   = (lane < 10) ? lane : 9;   // lanes 10..31 shadow lane 9

    float whh[4][10];
#pragma unroll
    for (int k = 0; k < 4; ++k)
#pragma unroll
        for (int j = 0; j < 10; ++j)
            whh[k][j] = W_hh[(size_t)(k * 10 + li) * 10 + j];

    float wih[4][10];
    float bias[4];
    if (FUSED) {
#pragma unroll
        for (int k = 0; k < 4; ++k) {
            bias[k] = b_ih[k * 10 + li] + b_hh[k * 10 + li];
#pragma unroll
            for (int j = 0; j < 10; ++j)
                wih[k][j] = W_ih[(size_t)(k * 10 + li) * 10 + j];
        }
    }

    float c  = c0[li];
    float hv = h0[li];
    float hs[10];
#pragma unroll
    for (int j = 0; j < 10; ++j) hs[j] = bcastf(hv, j);

    // One LSTM step from pre-activation gates (g0..g3 already include xg).
    auto step = [&](float g0, float g1, float g2, float g3) {
        g0 = dot10(hs, whh[0], g0);
        g1 = dot10(hs, whh[1], g1);
        g2 = dot10(hs, whh[2], g2);
        g3 = dot10(hs, whh[3], g3);
        float ig = fsigmoid(g0), fg = fsigmoid(g1);
        float gg = ftanhf(g2),   og = fsigmoid(g3);
        c = fmaf(fg, c, ig * gg);
        float h = og * ftanhf(c);
#pragma unroll
        for (int j = 0; j < 10; ++j) hs[j] = bcastf(h, j);
        hv = h;
    };

    constexpr int U = SCAN_PF;
    const int TU = (T / U) * U;

    if (!FUSED) {
        const float4* xg4 = (const float4*)src;   // [T*10] float4 (16B aligned)
        float4 buf[U];
#pragma unroll
        for (int u = 0; u < U; ++u) {             // prime pipeline
            int tn = (u < T) ? u : (T - 1);
            buf[u] = xg4[(size_t)tn * 10 + li];
        }
        for (int t = 0; t < TU; t += U) {
#pragma unroll
            for (int u = 0; u < U; ++u) {
                float4 g = buf[u];
                int tn = t + u + U; if (tn > T - 1) tn = T - 1;
                buf[u] = xg4[(size_t)tn * 10 + li];   // prefetch U ahead
                step(g.x, g.y, g.z, g.w);
            }
        }
#pragma unroll
        for (int u = 0; u < U; ++u) {             // unrolled guarded tail
            if (TU + u < T) {
                float4 g = buf[u];
                step(g.x, g.y, g.z, g.w);
            }
        }
    } else {
        float buf[U];
#pragma unroll
        for (int u = 0; u < U; ++u) {
            int tn = (u < T) ? u : (T - 1);
            buf[u] = src[(size_t)tn * 10 + li];
        }
        auto fstep = [&](float xv) {
            float xs[10];
#pragma unroll
            for (int j = 0; j < 10; ++j) xs[j] = bcastf(xv, j);
            float g0 = dot10(xs, wih[0], bias[0]);
            float g1 = dot10(xs, wih[1], bias[1]);
            float g2 = dot10(xs, wih[2], bias[2]);
            float g3 = dot10(xs, wih[3], bias[3]);
            step(g0, g1, g2, g3);
        };
        for (int t = 0; t < TU; t += U) {
#pragma unroll
            for (int u = 0; u < U; ++u) {
                float xv = buf[u];
                int tn = t + u + U; if (tn > T - 1) tn = T - 1;
                buf[u] = src[(size_t)tn * 10 + li];
                fstep(xv);
            }
        }
#pragma unroll
        for (int u = 0; u < U; ++u) {
            if (TU + u < T) fstep(buf[u]);
        }
    }

    if (lane < 10) out[lane] = hv;
}

// ---------------------------------------------------------------------------
extern "C" void kernel_launch(void* const* d_in, const int* in_sizes, int n_in,
                              void* d_out, int out_size, void* d_ws, size_t ws_size,
                              hipStream_t stream)
{
    const float* x   = (const float*)d_in[0];  // (T, 10)
    const float* h0  = (const float*)d_in[1];  // (1,1,10)
    const float* c0  = (const float*)d_in[2];  // (1,1,10)
    const float* Wih = (const float*)d_in[3];  // (40,10)
    const float* Whh = (const float*)d_in[4];  // (40,10)
    const float* bih = (const float*)d_in[5];  // (40,)
    const float* bhh = (const float*)d_in[6];  // (40,)
    float* out = (float*)d_out;

    const int H = 10;
    const int T = in_sizes[0] / H;

    size_t need = (size_t)T * 4 * H * sizeof(float);   // 160 MB for T=1e6
    if (d_ws && ws_size >= need) {
        float* xg = (float*)d_ws;
        int ntiles = (T + 15) / 16;
        int blocks = (ntiles + 3) / 4;                 // 4 waves / block
        xg_wmma_kernel<<<blocks, 128, 0, stream>>>(x, Wih, bih, bhh, xg, T, ntiles);
        lstm_scan_kernel<false><<<1, 32, 0, stream>>>(xg, Whh, nullptr, nullptr,
                                                      nullptr, h0, c0, out, T);
    } else {
        // Fallback: fused input-GEMM inside the scan (no workspace needed).
        lstm_scan_kernel<true><<<1, 32, 0, stream>>>(x, Whh, Wih, bih, bhh,
                                                     h0, c0, out, T);
    }
}